// CombinedLoss_80418967650486
// MI455X (gfx1250) — compile-verified
//
#include <hip/hip_runtime.h>
#include <hip/hip_bf16.h>
#include <math.h>

// Problem constants (from reference)
#define BATCH     8
#define HW        76800     // 240*320
#define NBIN      81        // bin edges per batch
#define NBIN_USED 82        // 81 edges + pad column
#define NBIN_PAD  96        // padded to 6*16 WMMA row-groups
#define NGROUPS   4800      // HW / 16 target columns per WMMA tile

// Workspace layout (floats / bit-patterns)
#define WS_GMAXT  0   // uint bits: global max of masked target
#define WS_GMAXB  1   // uint bits: global max of bin edges
#define WS_SUMD   2
#define WS_SUMD2  3
#define WS_SUMSQ  4
#define WS_CNT    5
#define WS_DIR2   8   // 8 floats, per-batch dir2 sums
#define WS_BINMIN 16  // 8*96 int (float bits), per-(batch,bin) running mins
#define WS_TOTAL  (WS_BINMIN + BATCH * NBIN_PAD)

typedef __attribute__((ext_vector_type(2))) float v2f;
typedef __attribute__((ext_vector_type(8))) float v8f;

__device__ __forceinline__ int imin2(int a, int b) { return a < b ? a : b; }

// ---------------------------------------------------------------- init
__global__ void k_init(float* wsF) {
    unsigned* wsU = (unsigned*)wsF;
    for (int i = threadIdx.x; i < WS_TOTAL; i += blockDim.x) {
        if (i < WS_BINMIN) {
            if (i == WS_GMAXT || i == WS_GMAXB) wsU[i] = 0u;     // bits of +0.0f
            else                                wsF[i] = 0.0f;
        } else {
            wsU[i] = 0x7F7FFFFFu;                                // FLT_MAX bits (== INT_MAX-ish)
        }
    }
}

// ---------------------------------------------------------------- global stats
__global__ void k_stats(const float* __restrict__ pred,
                        const float* __restrict__ tgt,
                        const float* __restrict__ bins,
                        const int*   __restrict__ mask,
                        float* wsF) {
    unsigned* wsU = (unsigned*)wsF;
    const int stride = gridDim.x * blockDim.x;
    float sd = 0.f, sd2 = 0.f, ssq = 0.f, c = 0.f, mx = 0.0f;
    for (int i = blockIdx.x * blockDim.x + threadIdx.x; i < BATCH * HW; i += stride) {
        int m = mask[i];
        float t = tgt[i], p = pred[i];
        if (m) {
            float d = __logf(p + 1e-10f) - __logf(t + 1e-10f);
            sd += d; sd2 += d * d;
            float e = p - t; ssq += e * e;
            c += 1.0f;
            mx = fmaxf(mx, t);
        }
    }
    // wave32 reduction
    for (int off = 16; off >= 1; off >>= 1) {
        sd  += __shfl_xor(sd,  off, 32);
        sd2 += __shfl_xor(sd2, off, 32);
        ssq += __shfl_xor(ssq, off, 32);
        c   += __shfl_xor(c,   off, 32);
        mx   = fmaxf(mx, __shfl_xor(mx, off, 32));
    }
    if ((threadIdx.x & 31) == 0) {
        atomicAdd(&wsF[WS_SUMD],  sd);
        atomicAdd(&wsF[WS_SUMD2], sd2);
        atomicAdd(&wsF[WS_SUMSQ], ssq);
        atomicAdd(&wsF[WS_CNT],   c);
        atomicMax(&wsU[WS_GMAXT], __float_as_uint(mx));   // non-negative: bit order == float order
    }
    if (blockIdx.x == 0) {
        for (int i = threadIdx.x; i < BATCH * NBIN; i += blockDim.x)
            atomicMax(&wsU[WS_GMAXB], __float_as_uint(bins[i]));
    }
}

// ---------------------------------------------------------------- fused chamfer (both directions)
// (a-b)^2 = [a^2, -2a, 1, 0] . [1, b, b^2, 0] -> one V_WMMA_F32_16X16X4_F32
// yields a 16x16 tile of pairwise squared distances (16 bins x 16 targets).
// All min-reductions done as signed-int min on float bit patterns: distances are
// >= 0 up to tiny rounding artifacts near 0, for which int order still picks the
// ~0 value. This avoids fminf's NaN-canonicalization VALU overhead.
__global__ void __launch_bounds__(256) k_chamfer(const float* __restrict__ tgt,
                                                 const float* __restrict__ bins,
                                                 const int*   __restrict__ mask,
                                                 float* wsF) {
    unsigned* wsU = (unsigned*)wsF;
    int*      wsI = (int*)wsF;
    __shared__ float s_bin[NBIN_PAD];
    const int BLK_PER_BATCH = 32;
    const int b          = blockIdx.x / BLK_PER_BATCH;
    const int blkInBatch = blockIdx.x % BLK_PER_BATCH;

    // pad value from global maxes (written by k_stats, stream-ordered)
    float gt  = __uint_as_float(wsU[WS_GMAXT]);
    float gb  = __uint_as_float(wsU[WS_GMAXB]);
    float mxv = fmaxf(gt, gb), mnv = fminf(gt, gb);
    float pad = mxv + (mxv - mnv) + 1.0f;

    for (int j = threadIdx.x; j < NBIN_PAD; j += blockDim.x)
        s_bin[j] = (j < NBIN) ? bins[b * NBIN + j] : ((j == NBIN) ? pad : 3.0e18f);
    __syncthreads();

    const int  lane = threadIdx.x & 31;
    const int  wave = threadIdx.x >> 5;
    const int  waveGlobal = blkInBatch * 8 + wave;      // uniform per wave
    const int  numWaves   = BLK_PER_BATCH * 8;
    const bool lo = (lane < 16);
    const int  n  = lane & 15;

    // A fragments: 16x4 f32; lanes 0-15 hold K=0,1 ; lanes 16-31 hold K=2,3
    v2f afrag[6];
#pragma unroll
    for (int i = 0; i < 6; ++i) {
        float a = s_bin[i * 16 + n];
        afrag[i][0] = lo ? a * a    : 1.0f;
        afrag[i][1] = lo ? -2.0f*a  : 0.0f;
    }
    const int BIGI = 0x7F7FFFFF;       // FLT_MAX bits
    int mtileI[6][8];
#pragma unroll
    for (int i = 0; i < 6; ++i)
#pragma unroll
        for (int r = 0; r < 8; ++r) mtileI[i][r] = BIGI;
    v8f czero = {};

    float d2acc = 0.0f;
    for (int g = waveGlobal; g < NGROUPS; g += numWaves) {
        int   idx = b * HW + g * 16 + n;
        float t   = mask[idx] ? tgt[idx] : pad;
        // B fragment: 4x16 f32; lanes 0-15 hold K=0,1 ; lanes 16-31 hold K=2,3
        v2f bfrag;
        bfrag[0] = lo ? 1.0f : t * t;
        bfrag[1] = lo ? t    : 0.0f;

        int colminI = BIGI;
#pragma unroll
        for (int i = 0; i < 6; ++i) {
            v8f d = __builtin_amdgcn_wmma_f32_16x16x4_f32(
                false, afrag[i], false, bfrag, (short)0, czero, false, false);
#pragma unroll
            for (int r = 0; r < 8; ++r) {
                int du = __float_as_int(d[r]);
                mtileI[i][r] = imin2(mtileI[i][r], du);   // dir1 running min over targets
                colminI      = imin2(colminI, du);        // dir2 min over this tile's rows
            }
        }
        colminI = imin2(colminI, __shfl_xor(colminI, 16, 32)); // combine row halves
        if (lo) d2acc += __int_as_float(colminI);              // one column per lane 0..15
    }
    for (int off = 16; off >= 1; off >>= 1) d2acc += __shfl_xor(d2acc, off, 32);
    if (lane == 0) atomicAdd(&wsF[WS_DIR2 + b], d2acc);

    // dir1: reduce running mins across the 16 lanes of each half, publish per-row
#pragma unroll
    for (int i = 0; i < 6; ++i) {
#pragma unroll
        for (int r = 0; r < 8; ++r) {
            int v = mtileI[i][r];
#pragma unroll
            for (int off = 1; off <= 8; off <<= 1)
                v = imin2(v, __shfl_xor(v, off, 32));     // stays within 16-lane half
            int row = i * 16 + r + (lo ? 0 : 8);
            if (lane == 0 || lane == 16)
                atomicMin(&wsI[WS_BINMIN + b * NBIN_PAD + row], v);
        }
    }
}

// ---------------------------------------------------------------- final combine
__global__ void k_final(const float* wsF, float* out) {
    const int* wsI = (const int*)wsF;
    __shared__ float sdata[256];
    float s = 0.0f;
    for (int k = threadIdx.x; k < BATCH * NBIN_USED; k += blockDim.x) {
        int b = k / NBIN_USED, j = k % NBIN_USED;
        s += __int_as_float(wsI[WS_BINMIN + b * NBIN_PAD + j]);  // dir1 sums
    }
    if (threadIdx.x < BATCH) s += wsF[WS_DIR2 + threadIdx.x];    // dir2 sums
    sdata[threadIdx.x] = s;
    __syncthreads();
    for (int off = 128; off >= 1; off >>= 1) {
        if ((int)threadIdx.x < off) sdata[threadIdx.x] += sdata[threadIdx.x + off];
        __syncthreads();
    }
    if (threadIdx.x == 0) {
        float chamfer = sdata[0] / (float)BATCH;                 // mean over batches
        float cnt = wsF[WS_CNT];
        float m1  = wsF[WS_SUMD]  / cnt;
        float m2  = wsF[WS_SUMD2] / cnt;
        float silog = 10.0f * sqrtf(m2 - 0.85f * m1 * m1);
        float l2    = sqrtf(wsF[WS_SUMSQ] / cnt);
        out[0] = l2 + silog + chamfer;
    }
}

// ---------------------------------------------------------------- launch
extern "C" void kernel_launch(void* const* d_in, const int* in_sizes, int n_in,
                              void* d_out, int out_size, void* d_ws, size_t ws_size,
                              hipStream_t stream) {
    const float* pred = (const float*)d_in[0];   // [8,1,240,320] f32
    const float* tgt  = (const float*)d_in[1];   // [8,1,240,320] f32
    const float* bins = (const float*)d_in[2];   // [8,81] f32
    const int*   mask = (const int*)d_in[3];     // [8,1,240,320] bool->int
    float* wsF = (float*)d_ws;
    float* out = (float*)d_out;

    k_init   <<<1,   256, 0, stream>>>(wsF);
    k_stats  <<<600, 256, 0, stream>>>(pred, tgt, bins, mask, wsF);
    k_chamfer<<<BATCH * 32, 256, 0, stream>>>(tgt, bins, mask, wsF);
    k_final  <<<1,   256, 0, stream>>>(wsF, out);
}